// VotingTextGCNModel_78984448573858
// MI455X (gfx1250) — compile-verified
//
#include <hip/hip_runtime.h>
#include <hip/hip_bf16.h>

// ---------------------------------------------------------------------------
// VotingTextGCN on MI455X (gfx1250, wave32)
//   layer1 GEMM  : V_WMMA_F32_16X16X4_F32, strip-mined: 1 wave = 16 rows x
//                  7 N-tiles (A fragment reused 7x -> A traffic ~306MB/branch)
//   aggregation  : wave-per-edge gather + global_atomic_add_f32 (L2-resident)
// ---------------------------------------------------------------------------

#define N_NODES 50000
#define FEAT    1536
#define KDIM    768
#define HID     200
#define NT_N    13           // ceil(200/16)
#define NT_M    3125         // 50000/16 exact
#define NGRP    2            // N-tile groups per strip (7 tiles each, padded)
#define GTILES  7

typedef float v2f __attribute__((ext_vector_type(2)));
typedef float v8f __attribute__((ext_vector_type(8)));

// ---------------- zero fill ----------------
__global__ void gcn_zero(float* __restrict__ p, long n) {
    long t = (long)blockIdx.x * blockDim.x + threadIdx.x;
    long stride = (long)gridDim.x * blockDim.x;
    for (; t < n; t += stride) p[t] = 0.0f;
}

// ---------------- layer-1 GEMM: Hpre[M,200] = X[:,off:off+768] @ W1 --------
// one wave per (16-row strip, group of 7 N-tiles); fp32 WMMA, K step 4
__global__ __launch_bounds__(128) void gcn_gemm1_wmma(
    const float* __restrict__ X, int col_off,
    const float* __restrict__ W, float* __restrict__ Hpre) {
    const int lane = threadIdx.x & 31;
    const int wave = threadIdx.x >> 5;
    const int sid  = blockIdx.x * 4 + wave;
    if (sid >= NT_M * NGRP) return;             // wave-uniform: EXEC stays full
    const int tm   = sid >> 1;                  // row strip
    const int g    = sid & 1;                   // N-tile group (tiles g*7 .. g*7+6)
    const int half = lane >> 4;                 // 0: K={0,1}, 1: K={2,3}
    const int l16  = lane & 15;

    const float* Arow = X + (size_t)(tm * 16 + l16) * FEAT + col_off;

    // per-lane B column for each of the 7 tiles (clamped for padding/edge)
    int bcol[GTILES];
    #pragma unroll
    for (int j = 0; j < GTILES; ++j) {
        int c = (g * GTILES + j) * 16 + l16;
        bcol[j] = c > HID - 1 ? HID - 1 : c;
    }

    v8f acc[GTILES];
    #pragma unroll
    for (int j = 0; j < GTILES; ++j) acc[j] = (v8f){};

    for (int k = 0; k < KDIM; k += 4) {
        const int ka = k + half * 2;
        const float2 av = *(const float2*)(Arow + ka);   // 8B aligned
        v2f a; a.x = av.x; a.y = av.y;
        const float* Wk0 = W + (size_t)ka * HID;
        const float* Wk1 = Wk0 + HID;
        #pragma unroll
        for (int j = 0; j < GTILES; ++j) {
            v2f b;
            b.x = Wk0[bcol[j]];
            b.y = Wk1[bcol[j]];
            acc[j] = __builtin_amdgcn_wmma_f32_16x16x4_f32(
                         false, a, false, b, (short)0, acc[j], false, false);
        }
    }

    const int rbase = tm * 16 + half * 8;       // C VGPR v -> rows v / v+8
    #pragma unroll
    for (int j = 0; j < GTILES; ++j) {
        const int ccol = (g * GTILES + j) * 16 + l16;
        if (ccol < HID) {
            #pragma unroll
            for (int v = 0; v < 8; ++v)
                Hpre[(size_t)(rbase + v) * HID + ccol] = acc[j][v];
        }
    }
}

// ---------------- edge scatter, 200 features: one wave per edge ------------
__global__ __launch_bounds__(256) void gcn_scatter200(
    const float* __restrict__ H, const int* __restrict__ src,
    const int* __restrict__ dst, const float* __restrict__ w,
    float* __restrict__ Agg, int n_edges) {
    const int lane = threadIdx.x & 31;
    const int e = (int)(((long)blockIdx.x * blockDim.x + threadIdx.x) >> 5);
    if (e >= n_edges) return;
    const int s = src[e], d = dst[e];
    const float we = w[e];
    const float4* hrow = (const float4*)(H + (size_t)s * HID);  // 800B rows, 16B aligned
    float* arow = Agg + (size_t)d * HID;
    for (int q = lane; q < HID / 4; q += 32) {
        float4 h = hrow[q];
        atomicAdd(arow + 4 * q + 0, h.x * we);
        atomicAdd(arow + 4 * q + 1, h.y * we);
        atomicAdd(arow + 4 * q + 2, h.z * we);
        atomicAdd(arow + 4 * q + 3, h.w * we);
    }
}

// ---------------- bias + relu in place -------------------------------------
__global__ void gcn_bias_relu(float* __restrict__ A, const float* __restrict__ b,
                              long total) {
    long t = (long)blockIdx.x * blockDim.x + threadIdx.x;
    long stride = (long)gridDim.x * blockDim.x;
    for (; t < total; t += stride) {
        float v = A[t] + b[(int)(t % HID)];
        A[t] = v > 0.0f ? v : 0.0f;
    }
}

// ---------------- layer-2 GEMM (200 -> 2): one wave per node ---------------
__global__ __launch_bounds__(256) void gcn_gemm2(
    const float* __restrict__ H, const float* __restrict__ W2,
    float* __restrict__ logits, int n_nodes) {
    const int lane = threadIdx.x & 31;
    const int node = (int)(((long)blockIdx.x * blockDim.x + threadIdx.x) >> 5);
    if (node >= n_nodes) return;
    const float* hrow = H + (size_t)node * HID;
    float a0 = 0.0f, a1 = 0.0f;
    for (int f = lane; f < HID; f += 32) {
        float h = hrow[f];
        a0 += h * W2[2 * f];
        a1 += h * W2[2 * f + 1];
    }
    #pragma unroll
    for (int off = 16; off > 0; off >>= 1) {
        a0 += __shfl_xor(a0, off, 32);
        a1 += __shfl_xor(a1, off, 32);
    }
    if (lane == 0) {
        logits[2 * (size_t)node]     = a0;
        logits[2 * (size_t)node + 1] = a1;
    }
}

// ---------------- edge scatter, 2 features: one thread per edge ------------
__global__ void gcn_scatter2(const float* __restrict__ logits,
                             const int* __restrict__ src,
                             const int* __restrict__ dst,
                             const float* __restrict__ w,
                             float* __restrict__ agg2, int n_edges) {
    int e = blockIdx.x * blockDim.x + threadIdx.x;
    if (e >= n_edges) return;
    int s = src[e], d = dst[e];
    float we = w[e];
    atomicAdd(&agg2[2 * (size_t)d],     logits[2 * (size_t)s]     * we);
    atomicAdd(&agg2[2 * (size_t)d + 1], logits[2 * (size_t)s + 1] * we);
}

// ---------------- bias + 2-class softmax -----------------------------------
__global__ void gcn_softmax2(const float* __restrict__ agg2,
                             const float* __restrict__ b2,
                             float* __restrict__ p, int n_nodes) {
    int i = blockIdx.x * blockDim.x + threadIdx.x;
    if (i >= n_nodes) return;
    float z0 = agg2[2 * (size_t)i]     + b2[0];
    float z1 = agg2[2 * (size_t)i + 1] + b2[1];
    float m = fmaxf(z0, z1);
    float e0 = expf(z0 - m), e1 = expf(z1 - m);
    float inv = 1.0f / (e0 + e1);
    p[2 * (size_t)i]     = e0 * inv;
    p[2 * (size_t)i + 1] = e1 * inv;
}

// ---------------- final max-vote + renormalize -----------------------------
__global__ void gcn_vote(const float* __restrict__ pa, const float* __restrict__ pb,
                         float* __restrict__ out, int n_nodes) {
    int i = blockIdx.x * blockDim.x + threadIdx.x;
    if (i >= n_nodes) return;
    float v0 = fmaxf(pa[2 * (size_t)i],     pb[2 * (size_t)i]);
    float v1 = fmaxf(pa[2 * (size_t)i + 1], pb[2 * (size_t)i + 1]);
    float inv = 1.0f / (v0 + v1);
    out[2 * (size_t)i]     = v0 * inv;
    out[2 * (size_t)i + 1] = v1 * inv;
}

// ---------------------------------------------------------------------------
extern "C" void kernel_launch(void* const* d_in, const int* in_sizes, int n_in,
                              void* d_out, int out_size, void* d_ws, size_t ws_size,
                              hipStream_t stream) {
    const float* x        = (const float*)d_in[0];
    const int*   edge_src = (const int*)d_in[1];
    const int*   edge_dst = (const int*)d_in[2];
    const float* edge_w   = (const float*)d_in[3];
    const int n_edges = in_sizes[1];

    // workspace layout (floats): Hpre[10M] | Agg[10M] | logits[100K] | agg2[100K] | p_a[100K] | p_b[100K]
    float* Hpre   = (float*)d_ws;
    float* Agg    = Hpre + (size_t)N_NODES * HID;
    float* logits = Agg  + (size_t)N_NODES * HID;
    float* agg2   = logits + (size_t)N_NODES * 2;
    float* p_a    = agg2   + (size_t)N_NODES * 2;
    float* p_b    = p_a    + (size_t)N_NODES * 2;

    const long hid_total = (long)N_NODES * HID;
    const int  gemm_blocks    = (NT_M * NGRP + 3) / 4;               // 4 waves/block
    const int  scatter_blocks = (int)(((long)n_edges * 32 + 255) / 256);
    const int  node_wave_blks = (int)(((long)N_NODES * 32 + 255) / 256);
    const int  node_blocks    = (N_NODES + 255) / 256;
    const int  edge_blocks    = (n_edges + 255) / 256;

    for (int br = 0; br < 2; ++br) {
        const float* W1 = (const float*)d_in[br ? 8 : 4];
        const float* b1 = (const float*)d_in[br ? 9 : 5];
        const float* W2 = (const float*)d_in[br ? 10 : 6];
        const float* b2 = (const float*)d_in[br ? 11 : 7];
        float* p = br ? p_b : p_a;
        const int col_off = br ? KDIM : 0;

        gcn_gemm1_wmma<<<gemm_blocks, 128, 0, stream>>>(x, col_off, W1, Hpre);
        gcn_zero<<<2048, 256, 0, stream>>>(Agg, hid_total);
        gcn_scatter200<<<scatter_blocks, 256, 0, stream>>>(Hpre, edge_src, edge_dst,
                                                           edge_w, Agg, n_edges);
        gcn_bias_relu<<<2048, 256, 0, stream>>>(Agg, b1, hid_total);
        gcn_gemm2<<<node_wave_blks, 256, 0, stream>>>(Agg, W2, logits, N_NODES);
        gcn_zero<<<512, 256, 0, stream>>>(agg2, (long)N_NODES * 2);
        gcn_scatter2<<<edge_blocks, 256, 0, stream>>>(logits, edge_src, edge_dst,
                                                      edge_w, agg2, n_edges);
        gcn_softmax2<<<node_blocks, 256, 0, stream>>>(agg2, b2, p, N_NODES);
    }
    gcn_vote<<<node_blocks, 256, 0, stream>>>(p_a, p_b, (float*)d_out, N_NODES);
}